// Decoder_69088843924187
// MI455X (gfx1250) — compile-verified
//
#include <hip/hip_runtime.h>

#define C 128
#define NL 4
#define NEG_SLOPE 0.2f
#define SA 132    // padded LDS stride (words) for activation tiles
#define SW 132    // padded LDS stride (words) for head weight matrix
#define WSTR 136  // padded weight row stride in halves (68 words): b128 hits each bank exactly 2x

typedef _Float16 h8v __attribute__((ext_vector_type(8)));   // 16B -> ds_load_b128
typedef float v2f __attribute__((ext_vector_type(2)));
typedef float v4f __attribute__((ext_vector_type(4)));
typedef float v8f __attribute__((ext_vector_type(8)));

// ---------------------------------------------------------------------------
// Phase 1: serial RNN recurrence. One workgroup, 256 threads (8 waves,
// 2/SIMD so LDS and VALU pipes overlap across waves).
// All 8 weight matrices LDS-resident as f16 with bank-conflict-free padded
// rows; f32 accumulate via v_fma_mix. Channel ch is computed by two threads
// (K split in halves) with an LDS partial-sum reduce.
// Hidden state is double-buffered by step parity with one slot per layer,
// so each layer needs no write-after-read fencing (2 barriers/layer).
// ---------------------------------------------------------------------------
__global__ __launch_bounds__(256, 1) void rnn_recurrence(
    const float* __restrict__ x, const float* __restrict__ Wih,
    const float* __restrict__ Whh, const float* __restrict__ bih,
    const float* __restrict__ bhh, const int* __restrict__ max_steps,
    float* __restrict__ enc)
{
    __shared__ _Float16 swih[NL * C * WSTR];   // ~136 KB
    __shared__ _Float16 swhh[NL * C * WSTR];   // ~136 KB
    __shared__ float    sb[NL * C];            // bih + bhh combined
    __shared__ float    hid2[2][NL * C];       // [parity][layer][channel]
    __shared__ float    part[2][C];            // K-half partial sums

    const int tid  = threadIdx.x;
    const int ch   = tid & (C - 1);
    const int half = tid >> 7;                 // 0: K=0..63, 1: K=64..127

    for (int i = tid; i < NL * C * C; i += 256) {
        int r = i >> 7, c = i & (C - 1);
        swih[r * WSTR + c] = (_Float16)Wih[i];
        swhh[r * WSTR + c] = (_Float16)Whh[i];
    }
    for (int i = tid; i < NL * C; i += 256) {
        sb[i]      = bih[i] + bhh[i];
        hid2[0][i] = 0.0f;
        hid2[1][i] = (i < C) ? x[i] : 0.0f;    // hid[layer 0] = conditioning x
    }
    __syncthreads();

    const int T = max_steps[0];
    for (int t = 0; t < T; ++t) {
        const int p = t & 1, q = p ^ 1;
        for (int l = 0; l < NL; ++l) {
            // layer input: top-layer output of previous step (l==0) or the
            // slot just written this step (l>0); h_prev: this layer, prev step
            const float* in = (l == 0) ? &hid2[q][(NL - 1) * C]
                                       : &hid2[p][(l - 1) * C];
            const float* hp = &hid2[q][l * C];
            const v4f* in4 = (const v4f*)in;   // wave-uniform (broadcast) reads
            const v4f* hp4 = (const v4f*)hp;

            const h8v* wi8 = (const h8v*)(swih + (l * C + ch) * WSTR + half * 64);
            const h8v* wh8 = (const h8v*)(swhh + (l * C + ch) * WSTR + half * 64);

            float acc = 0.0f;
            #pragma unroll
            for (int k8 = 0; k8 < 8; ++k8) {   // 8 halves per b128, 64 K per thread
                h8v wa = wi8[k8];
                h8v wb = wh8[k8];
                v4f i0 = in4[half * 16 + k8 * 2];
                v4f i1 = in4[half * 16 + k8 * 2 + 1];
                v4f h0 = hp4[half * 16 + k8 * 2];
                v4f h1 = hp4[half * 16 + k8 * 2 + 1];
                acc += (float)wa[0] * i0.x + (float)wa[1] * i0.y;
                acc += (float)wa[2] * i0.z + (float)wa[3] * i0.w;
                acc += (float)wa[4] * i1.x + (float)wa[5] * i1.y;
                acc += (float)wa[6] * i1.z + (float)wa[7] * i1.w;
                acc += (float)wb[0] * h0.x + (float)wb[1] * h0.y;
                acc += (float)wb[2] * h0.z + (float)wb[3] * h0.w;
                acc += (float)wb[4] * h1.x + (float)wb[5] * h1.y;
                acc += (float)wb[6] * h1.z + (float)wb[7] * h1.w;
            }
            part[half][ch] = acc;
            __syncthreads();
            if (half == 0) {                   // waves 0-3: reduce + publish
                float s = sb[l * C + ch] + part[0][ch] + part[1][ch];
                s = s > 0.0f ? s : 0.0f;       // relu
                hid2[p][l * C + ch] = s;
                if (l == NL - 1) enc[t * C + ch] = s;
            }
            __syncthreads();
        }
    }
}

// ---------------------------------------------------------------------------
// Phase 2: head MLP over encodings using V_WMMA_F32_16X16X4_F32 (full f32).
// Block = 128 threads (4 waves); each wave owns a 16-row tile, block = 64 rows.
// (unchanged from round 0 — instruction mix verified good)
// ---------------------------------------------------------------------------
__global__ __launch_bounds__(128, 1) void head_kernel(
    const float* __restrict__ enc,      // [T][C]
    const float* __restrict__ rb_w,     // [2][2][C][C] (out, in)
    const float* __restrict__ rb_b,     // [2][2][C]
    const float* __restrict__ out_w,    // [od][C]
    const float* __restrict__ out_b,    // [od]
    float* __restrict__ out,            // [T][10]
    int od, int col0)
{
    __shared__ float wlds[C * SW];
    __shared__ float blds[C];
    __shared__ float xbuf[4][16 * SA];
    __shared__ float hbuf[4][16 * SA];

    const int tid  = threadIdx.x;
    const int wave = tid >> 5;
    const int lane = tid & 31;
    const int ln   = lane & 15;
    const int hi   = lane >> 4;
    const int row0 = blockIdx.x * 64 + wave * 16;

    for (int i = lane; i < 16 * C; i += 32) {
        int m = i >> 7, c = i & (C - 1);
        xbuf[wave][m * SA + c] = enc[(row0 + m) * C + c];
    }

    for (int blk = 0; blk < 2; ++blk) {
        for (int ly = 0; ly < 2; ++ly) {
            const float* W = rb_w + (blk * 2 + ly) * C * C;
            const float* B = rb_b + (blk * 2 + ly) * C;
            __syncthreads();
            for (int i = tid; i < C * C; i += 128)
                wlds[(i >> 7) * SW + (i & (C - 1))] = W[i];
            if (tid < C) blds[tid] = B[tid];
            __syncthreads();

            const float* src = (ly == 0) ? xbuf[wave] : hbuf[wave];
            float*       dst = (ly == 0) ? hbuf[wave] : xbuf[wave];

            v8f acc[8];
            #pragma unroll
            for (int nt = 0; nt < 8; ++nt) {
                float bv = blds[nt * 16 + ln];
                #pragma unroll
                for (int r = 0; r < 8; ++r) {
                    float cv = bv;
                    if (ly == 1)
                        cv += xbuf[wave][(r + 8 * hi) * SA + nt * 16 + ln];
                    acc[nt][r] = cv;
                }
            }

            for (int k = 0; k < 32; ++k) {
                v2f a;
                a.x = src[ln * SA + 4 * k + 2 * hi];
                a.y = src[ln * SA + 4 * k + 2 * hi + 1];
                #pragma unroll
                for (int nt = 0; nt < 8; ++nt) {
                    v2f b;
                    b.x = wlds[(nt * 16 + ln) * SW + 4 * k + 2 * hi];
                    b.y = wlds[(nt * 16 + ln) * SW + 4 * k + 2 * hi + 1];
                    acc[nt] = __builtin_amdgcn_wmma_f32_16x16x4_f32(
                        false, a, false, b, (short)0, acc[nt], false, false);
                }
            }

            #pragma unroll
            for (int nt = 0; nt < 8; ++nt) {
                #pragma unroll
                for (int r = 0; r < 8; ++r) {
                    float v = acc[nt][r];
                    v = v > 0.0f ? v : NEG_SLOPE * v;
                    dst[(r + 8 * hi) * SA + nt * 16 + ln] = v;
                }
            }
        }
    }
    __syncthreads();

    for (int o = hi; o < od; o += 2) {
        const float* wo = out_w + o * C;
        float s = out_b[o];
        for (int c = 0; c < C; ++c) s += xbuf[wave][ln * SA + c] * wo[c];
        out[(row0 + ln) * 10 + col0 + o] = s;
    }
}

// ---------------------------------------------------------------------------
extern "C" void kernel_launch(void* const* d_in, const int* in_sizes, int n_in,
                              void* d_out, int out_size, void* d_ws, size_t ws_size,
                              hipStream_t stream)
{
    const float* x         = (const float*)d_in[0];
    const float* W_ih      = (const float*)d_in[1];
    const float* W_hh      = (const float*)d_in[2];
    const float* b_ih      = (const float*)d_in[3];
    const float* b_hh      = (const float*)d_in[4];
    const float* atom_rb_w = (const float*)d_in[5];
    const float* atom_rb_b = (const float*)d_in[6];
    const float* atom_w    = (const float*)d_in[7];
    const float* atom_b    = (const float*)d_in[8];
    const float* pos_rb_w  = (const float*)d_in[9];
    const float* pos_rb_b  = (const float*)d_in[10];
    const float* pos_w     = (const float*)d_in[11];
    const float* pos_b     = (const float*)d_in[12];
    const float* mag_rb_w  = (const float*)d_in[13];
    const float* mag_rb_b  = (const float*)d_in[14];
    const float* mag_w     = (const float*)d_in[15];
    const float* mag_b     = (const float*)d_in[16];
    const int*   max_steps = (const int*)d_in[17];

    float* out = (float*)d_out;
    float* enc = (float*)d_ws;              // [T][128] f32 encodings (16 MB)
    const int T = out_size / 10;            // e.g. 32768

    rnn_recurrence<<<1, 256, 0, stream>>>(x, W_ih, W_hh, b_ih, b_hh, max_steps, enc);

    const int blocks = T / 64;              // T is a multiple of 64
    head_kernel<<<blocks, 128, 0, stream>>>(enc, atom_rb_w, atom_rb_b, atom_w, atom_b, out, 8, 0);
    head_kernel<<<blocks, 128, 0, stream>>>(enc, pos_rb_w,  pos_rb_b,  pos_w,  pos_b,  out, 1, 8);
    head_kernel<<<blocks, 128, 0, stream>>>(enc, mag_rb_w,  mag_rb_b,  mag_w,  mag_b,  out, 1, 9);
}